// DynamicVoxelization_49048526520404
// MI455X (gfx1250) — compile-verified
//
#include <hip/hip_runtime.h>
#include <hip/hip_bf16.h>

typedef unsigned int u32;
typedef __attribute__((ext_vector_type(2))) float v2f;
typedef __attribute__((ext_vector_type(8))) float v8f;

// ---- compile-time module config (matches reference) ----
#define GX 352
#define GY 400
#define GZ 10
#define GRID_G (GZ * GY * GX)          // 1,408,000
#define NBATCH 2
#define SENT ((u32)(NBATCH * GRID_G))  // 2,816,000
#define NKEYS (SENT + 1u)              // bins incl. sentinel
#define CF 64                          // feature dim
#define SCAN_TPB 256
#define SCAN_ITEMS 4
#define SCAN_BINS (SCAN_TPB * SCAN_ITEMS) // 1024 bins per scan block

// ---------------- zero-init ----------------
__global__ void zero_u32_kernel(u32* p, long long n) {
    long long i = (long long)blockIdx.x * blockDim.x + threadIdx.x;
    if (i < n) p[i] = 0u;
}
__global__ void zero_f32_kernel(float* p, long long n) {
    long long i = (long long)blockIdx.x * blockDim.x + threadIdx.x;
    if (i < n) p[i] = 0.0f;
}

// ---------------- 1) key per point + histogram ----------------
__global__ void keys_hist_kernel(const float* __restrict__ pts,
                                 u32* __restrict__ keys,
                                 u32* __restrict__ cnt, int N) {
    int i = blockIdx.x * blockDim.x + threadIdx.x;
    if (i >= N) return;
    float4 p = ((const float4*)pts)[i];   // global_load_b128
    // f32 math then trunc-toward-zero cast, exactly like the reference
    long long cz = (long long)((p.w - (-3.0f))  / 0.4f);
    long long cy = (long long)((p.z - (-40.0f)) / 0.2f);
    long long cx = (long long)((p.y - 0.0f)     / 0.2f);
    bool keep = (cz >= 0) && (cz < GZ) && (cy >= 0) && (cy < GY) &&
                (cx >= 0) && (cx < GX);
    u32 key = SENT;
    if (keep) {
        long long bi = (long long)p.x;
        key = (u32)(((bi * GZ + cz) * GY + cy) * GX + cx);
    }
    keys[i] = key;
    atomicAdd(&cnt[key], 1u);   // non-returning GLOBAL_ATOMIC_ADD_U32
}

// ---------------- 2) per-block flag scan over bins ----------------
__global__ void scan_block_kernel(const u32* __restrict__ cnt,
                                  u32* __restrict__ rank,
                                  u32* __restrict__ bsums, int K) {
    __shared__ u32 s[SCAN_TPB];
    int base = blockIdx.x * SCAN_BINS + threadIdx.x * SCAN_ITEMS;
    u32 f[SCAN_ITEMS];
    u32 tsum = 0;
#pragma unroll
    for (int j = 0; j < SCAN_ITEMS; ++j) {
        int i = base + j;
        u32 fl = (i < K && cnt[i] > 0u) ? 1u : 0u;
        f[j] = fl;
        tsum += fl;
    }
    s[threadIdx.x] = tsum;
    __syncthreads();
    for (int off = 1; off < SCAN_TPB; off <<= 1) {
        u32 v = (threadIdx.x >= (unsigned)off) ? s[threadIdx.x - off] : 0u;
        __syncthreads();
        s[threadIdx.x] += v;
        __syncthreads();
    }
    u32 run = s[threadIdx.x] - tsum;  // exclusive prefix of this thread
#pragma unroll
    for (int j = 0; j < SCAN_ITEMS; ++j) {
        int i = base + j;
        if (i < K) rank[i] = run;
        run += f[j];
    }
    if (threadIdx.x == SCAN_TPB - 1) bsums[blockIdx.x] = s[SCAN_TPB - 1];
}

// ---------------- 3) serial scan of block sums (tiny: ~2.8k) ----------------
__global__ void scan_sums_kernel(u32* __restrict__ bsums,
                                 u32* __restrict__ totalU, int nb) {
    if (blockIdx.x != 0 || threadIdx.x != 0) return;
    u32 run = 0;
    for (int i = 0; i < nb; ++i) {
        u32 t = bsums[i];
        bsums[i] = run;
        run += t;
    }
    totalU[0] = run;  // number of unique keys (incl. SENT)
}

// ---------------- 4) add offsets, emit sorted unique keys ----------------
__global__ void scan_add_kernel(const u32* __restrict__ cnt,
                                u32* __restrict__ rank,
                                const u32* __restrict__ boff,
                                u32* __restrict__ uniq, int K) {
    int i = blockIdx.x * blockDim.x + threadIdx.x;
    if (i >= K) return;
    u32 r = rank[i] + boff[i / SCAN_BINS];
    rank[i] = r;
    if (cnt[i] > 0u) uniq[r] = (u32)i;
}

// ---------------- 5) scatter-add features into sorted voxel rows ----------------
__global__ void scatter_kernel(const float* __restrict__ feat,
                               const u32* __restrict__ keys,
                               const u32* __restrict__ rank,
                               float* __restrict__ vox, int N) {
    long long t = (long long)blockIdx.x * blockDim.x + threadIdx.x;
    long long total = (long long)N * CF;
    if (t >= total) return;
    // stream-prefetch far ahead (global_prefetch_b8)
    long long pf = t + 16384;
    if (pf < total) __builtin_prefetch(&feat[pf], 0, 1);
    int i = (int)(t >> 6);
    int c = (int)(t & 63);
    u32 key = keys[i];
    if (key >= SENT) return;  // SENT segment is zeroed in the reference anyway
    u32 r = rank[key];
    atomicAdd(&vox[(size_t)r * CF + c], feat[t]);  // GLOBAL_ATOMIC_ADD_F32
}

// ---------------- 6) scale rows by 1/count via chained WMMA ----------------
// Each wave owns one 16x16 tile of the [Rtot, 64] sum matrix and computes
// D = diag(inv) * S as 4 accumulating V_WMMA_F32_16X16X4_F32:
//   group g: B = rows S[rbase+4g .. rbase+4g+3][cbase..cbase+15]
//            A[m][k] = (m == 4g+k) ? inv[m] : 0
// Wave-uniform fast path (full tiles): branch-free loads/stores with
// immediate offsets off a single tile base pointer. EXEC all-1s in both
// paths (WMMA requirement).
__global__ void finalize_wmma_kernel(float* __restrict__ vox,
                                     const u32* __restrict__ uniq,
                                     const u32* __restrict__ cnt,
                                     const u32* __restrict__ totalU,
                                     int Rtot) {
    int wave = (int)((blockIdx.x * (unsigned)blockDim.x + threadIdx.x) >> 5);
    int lane = threadIdx.x & 31;
    int nRow16 = (Rtot + 15) >> 4;
    int nTiles = nRow16 * 4;       // 4 column tiles of 16 across CF=64
    if (wave >= nTiles) return;    // uniform per wave
    int rg = wave >> 2, ct = wave & 3;
    int rbase = rg << 4, cbase = ct << 4;
    int half = lane >> 4, lm = lane & 15;

    u32 U = totalU[0];
    int mrow = rbase + lm;         // row this lane scales (lanes 16-31 mirror)
    float* tp = vox + (size_t)rbase * CF + cbase + lm;  // tile base, this lane's col

    if (rbase + 16 <= Rtot) {
        // ---- fast path: full 16-row tile, no guards, no clamps ----
        float inv = 0.0f;
        if ((u32)mrow < U) {
            u32 key = uniq[mrow];
            if (key < SENT) {
                u32 c = cnt[key];
                inv = 1.0f / (float)(c > 0u ? c : 1u);
            }
        }
        v8f acc = {};
#pragma unroll
        for (int g = 0; g < 4; ++g) {
            int k0 = (g << 2) + (half << 1);   // K index handled by VGPR .x
            v2f a, b;
            a.x = (lm == k0)     ? inv : 0.0f;
            a.y = (lm == k0 + 1) ? inv : 0.0f;
            b.x = tp[(size_t)k0 * CF];         // row rbase+k0,   col cbase+lm
            b.y = tp[(size_t)(k0 + 1) * CF];   // row rbase+k0+1
            acc = __builtin_amdgcn_wmma_f32_16x16x4_f32(
                false, a, false, b, (short)0, acc, false, false);
        }
#pragma unroll
        for (int j = 0; j < 8; ++j)            // D: VGPR j -> row j (+8 for hi half)
            tp[(size_t)(j + (half << 3)) * CF] = acc[j];
    } else {
        // ---- guarded path: only the single last (partial) row group ----
        float inv = 0.0f;
        if (mrow < Rtot && (u32)mrow < U) {
            u32 key = uniq[mrow];
            if (key < SENT) {
                u32 c = cnt[key];
                inv = 1.0f / (float)(c > 0u ? c : 1u);
            }
        }
        v8f acc = {};
        int maxRow = Rtot - 1;
#pragma unroll
        for (int g = 0; g < 4; ++g) {
            int k0 = (g << 2) + (half << 1);
            v2f a, b;
            a.x = (lm == k0)     ? inv : 0.0f;
            a.y = (lm == k0 + 1) ? inv : 0.0f;
            int r0 = rbase + k0, r1 = r0 + 1;
            r0 = r0 > maxRow ? maxRow : r0;    // clamped rows multiply by a==0
            r1 = r1 > maxRow ? maxRow : r1;
            b.x = vox[(size_t)r0 * CF + cbase + lm];
            b.y = vox[(size_t)r1 * CF + cbase + lm];
            acc = __builtin_amdgcn_wmma_f32_16x16x4_f32(
                false, a, false, b, (short)0, acc, false, false);
        }
#pragma unroll
        for (int j = 0; j < 8; ++j) {
            int row = rbase + j + (half << 3);
            if (row < Rtot) vox[(size_t)row * CF + cbase + lm] = acc[j];
        }
    }
}

// ---------------- 7) decode coors [b,z,y,x] (floats; -1 for padding) ----------------
__global__ void coors_kernel(const u32* __restrict__ uniq,
                             const u32* __restrict__ totalU,
                             float* __restrict__ out, int Rtot) {
    int r = blockIdx.x * blockDim.x + threadIdx.x;
    if (r >= Rtot) return;
    u32 U = totalU[0];
    float4 c4 = make_float4(-1.0f, -1.0f, -1.0f, -1.0f);
    if ((u32)r < U) {
        u32 key = uniq[r];
        if (key < SENT) {
            u32 bb = key / (u32)GRID_G;
            u32 rem = key % (u32)GRID_G;
            u32 zz = rem / (u32)(GY * GX);
            u32 yy = (rem / (u32)GX) % (u32)GY;
            u32 xx = rem % (u32)GX;
            c4 = make_float4((float)bb, (float)zz, (float)yy, (float)xx);
        }
    }
    ((float4*)out)[r] = c4;   // global_store_b128
}

static inline size_t align256(size_t x) { return (x + 255) & ~(size_t)255; }

extern "C" void kernel_launch(void* const* d_in, const int* in_sizes, int n_in,
                              void* d_out, int out_size, void* d_ws, size_t ws_size,
                              hipStream_t stream) {
    const float* points   = (const float*)d_in[0];
    const float* features = (const float*)d_in[1];
    const int N = in_sizes[0] / 4;
    const int Rtot = N + 1;

    float* vox   = (float*)d_out;                      // [Rtot, 64]
    float* coors = (float*)d_out + (size_t)Rtot * CF;  // [Rtot, 4]

    // workspace partition (~26 MB)
    char* w = (char*)d_ws;
    u32* keys = (u32*)w;  w += align256((size_t)N * 4);
    u32* cnt  = (u32*)w;  w += align256((size_t)NKEYS * 4);
    u32* rank = (u32*)w;  w += align256((size_t)NKEYS * 4);
    const int nb = (int)((NKEYS + SCAN_BINS - 1) / SCAN_BINS);
    u32* bsums  = (u32*)w; w += align256((size_t)(nb + 1) * 4);
    u32* uniq   = (u32*)w; w += align256((size_t)Rtot * 4);
    u32* totalU = (u32*)w; w += align256(4);
    (void)ws_size; (void)n_in; (void)out_size;

    const int TPB = 256;

    // zero counts + output sums (deterministic every call)
    {
        long long n = (long long)NKEYS;
        zero_u32_kernel<<<(unsigned)((n + TPB - 1) / TPB), TPB, 0, stream>>>(cnt, n);
        long long m = (long long)Rtot * CF;
        zero_f32_kernel<<<(unsigned)((m + TPB - 1) / TPB), TPB, 0, stream>>>(vox, m);
    }

    keys_hist_kernel<<<(N + TPB - 1) / TPB, TPB, 0, stream>>>(points, keys, cnt, N);

    scan_block_kernel<<<nb, SCAN_TPB, 0, stream>>>(cnt, rank, bsums, (int)NKEYS);
    scan_sums_kernel<<<1, 1, 0, stream>>>(bsums, totalU, nb);
    scan_add_kernel<<<(int)((NKEYS + TPB - 1) / TPB), TPB, 0, stream>>>(
        cnt, rank, bsums, uniq, (int)NKEYS);

    {
        long long total = (long long)N * CF;
        scatter_kernel<<<(unsigned)((total + TPB - 1) / TPB), TPB, 0, stream>>>(
            features, keys, rank, vox, N);
    }

    {
        int nTiles = ((Rtot + 15) >> 4) * 4;
        long long threads = (long long)nTiles * 32;
        finalize_wmma_kernel<<<(unsigned)((threads + TPB - 1) / TPB), TPB, 0, stream>>>(
            vox, uniq, cnt, totalU, Rtot);
    }

    coors_kernel<<<(Rtot + TPB - 1) / TPB, TPB, 0, stream>>>(uniq, totalU, coors, Rtot);
}